// GCN_55353538511629
// MI455X (gfx1250) — compile-verified
//
#include <hip/hip_runtime.h>
#include <hip/hip_bf16.h>

typedef __attribute__((ext_vector_type(2))) float v2f;
typedef __attribute__((ext_vector_type(8))) float v8f;

#define HDIM 128

// ---------------------------------------------------------------------------
// Fused GEMM + self-loop epilogue:
//   xw  = act(A) @ W                      (act = optional ReLU)
//   out = deg_inv[:,None] * xw + bias     (edge scatter then atomically adds)
//
// Block = 256 threads (8 waves) covering 64 rows x 128 cols.
// A-slab (64 x K) staged into LDS with CDNA5 async global->LDS copies
// (ASYNCcnt path), row stride padded to K+4 floats ((K+4)/4 odd) so the
// float2 A-fragment reads are 64-bank conflict-free for both half-waves.
// Each wave owns one 16-col tile and 4 M-tiles; B-fragment (from W, L2
// resident) is loaded once per k-step and reused across 4 WMMAs.
// ---------------------------------------------------------------------------
template <int K, bool RELU_IN>
__global__ __launch_bounds__(256) void gcn_gemm_fused(
    const float* __restrict__ A, const float* __restrict__ W,
    const float* __restrict__ bias, const float* __restrict__ deginv,
    float* __restrict__ xw, float* __restrict__ out, int nrows) {
  constexpr int KP  = K + 4;      // padded LDS row stride (floats)
  constexpr int CPR = K / 4;      // float4 chunks per row
  __shared__ float A_s[64 * KP];

  const int lane = threadIdx.x & 31;
  const int wave = threadIdx.x >> 5;      // 8 waves -> 8 col tiles (128/16)
  const int row0 = blockIdx.x << 6;       // 64 rows per block
  const int col0 = wave << 4;
  const int n    = lane & 15;             // == m for A-frag
  const int koff = (lane >> 4) << 1;      // 0 or 2

  // ---- async copy of the 64 x K A-slab into LDS (16 B per lane per iter) ----
  for (int i = threadIdx.x; i < 64 * CPR; i += 256) {
    const int r = i / CPR;
    const int c = i % CPR;
    int gr = row0 + r;
    if (gr > nrows - 1) gr = nrows - 1;   // clamp (EXEC stays all-1s)
    const float* gp = A + (size_t)gr * K + c * 4;
    unsigned lo = (unsigned)(size_t)(const void*)&A_s[r * KP + c * 4];
    asm volatile("global_load_async_to_lds_b128 %0, %1, off"
                 :: "v"(lo), "v"(gp) : "memory");
  }
  asm volatile("s_wait_asynccnt 0x0" ::: "memory");
  __syncthreads();

  // ---- WMMA main loop: 4 M-tiles per wave, B-frag reused across tiles ----
  v8f acc[4] = {};
  const float* Wc = W + col0 + n;
#pragma unroll 8
  for (int k = 0; k < K; k += 4) {
    v2f b;
    b.x = Wc[(size_t)(k + koff) * HDIM];
    b.y = Wc[(size_t)(k + koff + 1) * HDIM];
#pragma unroll
    for (int t = 0; t < 4; ++t) {
      const float* ap = &A_s[(t * 16 + n) * KP + k + koff];
      float a0 = ap[0], a1 = ap[1];
      if (RELU_IN) { a0 = fmaxf(a0, 0.0f); a1 = fmaxf(a1, 0.0f); }
      v2f a; a.x = a0; a.y = a1;
      // 8 args: (neg_a, A, neg_b, B, c_mod, C, reuse_a, reuse_b)
      acc[t] = __builtin_amdgcn_wmma_f32_16x16x4_f32(false, a, false, b,
                                                     (short)0, acc[t],
                                                     false, false);
    }
  }

  // ---- fused epilogue: write xw and out = deg_inv*xw + bias ----
  const int mb = (lane >> 4) << 3;        // 0 or 8
  const float bn = bias[col0 + n];
  if (row0 + 64 <= nrows) {               // uniform fast path: no guards
#pragma unroll
    for (int t = 0; t < 4; ++t)
#pragma unroll
      for (int j = 0; j < 8; ++j) {
        const int grow = row0 + t * 16 + mb + j;
        const size_t off = (size_t)grow * HDIM + col0 + n;
        const float v = acc[t][j];
        xw[off] = v;
        out[off] = deginv[grow] * v + bn;
      }
  } else {                                // tail block
#pragma unroll
    for (int t = 0; t < 4; ++t)
#pragma unroll
      for (int j = 0; j < 8; ++j) {
        const int grow = row0 + t * 16 + mb + j;
        if (grow < nrows) {
          const size_t off = (size_t)grow * HDIM + col0 + n;
          const float v = acc[t][j];
          xw[off] = v;
          out[off] = deginv[grow] * v + bn;
        }
      }
  }
}

// ---------------------------------------------------------------------------
// Degree / norm precompute
// ---------------------------------------------------------------------------
__global__ void init_deg(float* __restrict__ deg, int n) {
  int i = blockIdx.x * blockDim.x + threadIdx.x;
  if (i < n) deg[i] = 1.0f;               // self-loop included
}

__global__ void accum_deg(const int* __restrict__ col, float* __restrict__ deg,
                          int e) {
  int i = blockIdx.x * blockDim.x + threadIdx.x;
  if (i < e) atomicAdd(&deg[col[i]], 1.0f);
}

__global__ void finalize_deg(const float* __restrict__ deg,
                             float* __restrict__ disqrt,
                             float* __restrict__ deginv, int n) {
  int i = blockIdx.x * blockDim.x + threadIdx.x;
  if (i < n) {
    float d = deg[i];
    disqrt[i] = rsqrtf(d);
    deginv[i] = 1.0f / d;
  }
}

__global__ void edge_norm(const int* __restrict__ row,
                          const int* __restrict__ col,
                          const float* __restrict__ disqrt,
                          float* __restrict__ nrm, int e) {
  int i = blockIdx.x * blockDim.x + threadIdx.x;
  if (i < e) nrm[i] = disqrt[row[i]] * disqrt[col[i]];
}

// ---------------------------------------------------------------------------
// Edge scatter: one wave per edge. Gather xw[row[e]] (float4/lane, L2-resident
// since xw = 51.2 MB < 192 MB L2), scale by norm, atomic-add into out[col[e]].
// ---------------------------------------------------------------------------
__global__ __launch_bounds__(256) void edge_scatter(
    const float* __restrict__ xw, const int* __restrict__ row,
    const int* __restrict__ col, const float* __restrict__ nrm,
    float* __restrict__ out, int e) {
  int edge = blockIdx.x * 8 + (threadIdx.x >> 5);
  if (edge >= e) return;
  int lane = threadIdx.x & 31;
  int src = row[edge];
  int dst = col[edge];
  float nv = nrm[edge];
  float4 v = ((const float4*)(xw + (size_t)src * HDIM))[lane];
  float* d = out + (size_t)dst * HDIM + lane * 4;
  atomicAdd(d + 0, nv * v.x);
  atomicAdd(d + 1, nv * v.y);
  atomicAdd(d + 2, nv * v.z);
  atomicAdd(d + 3, nv * v.w);
}

// ---------------------------------------------------------------------------
// Pooling
// ---------------------------------------------------------------------------
__global__ void zero_f32(float* __restrict__ p, int n) {
  int i = blockIdx.x * blockDim.x + threadIdx.x;
  if (i < n) p[i] = 0.0f;
}

__global__ __launch_bounds__(256) void pool_accum(
    const float* __restrict__ h, const int* __restrict__ batch,
    float* __restrict__ sums, float* __restrict__ counts, int n) {
  int node = blockIdx.x * 8 + (threadIdx.x >> 5);
  if (node >= n) return;
  int lane = threadIdx.x & 31;
  int g = batch[node];
  float4 v = ((const float4*)(h + (size_t)node * HDIM))[lane];
  float* d = sums + (size_t)g * HDIM + lane * 4;
  atomicAdd(d + 0, v.x);
  atomicAdd(d + 1, v.y);
  atomicAdd(d + 2, v.z);
  atomicAdd(d + 3, v.w);
  if (lane == 0) atomicAdd(&counts[g], 1.0f);
}

__global__ __launch_bounds__(256) void final_pool(
    const float* __restrict__ sums, const float* __restrict__ counts,
    const float* __restrict__ Wlin, const float* __restrict__ blin,
    float* __restrict__ out, int G) {
  int g = blockIdx.x * 8 + (threadIdx.x >> 5);
  if (g >= G) return;
  int lane = threadIdx.x & 31;
  float4 s = ((const float4*)(sums + (size_t)g * HDIM))[lane];
  float4 w = ((const float4*)Wlin)[lane];
  float p = s.x * w.x + s.y * w.y + s.z * w.z + s.w * w.w;
#pragma unroll
  for (int off = 16; off > 0; off >>= 1) p += __shfl_down(p, off, 32);
  if (lane == 0) out[g] = p / fmaxf(counts[g], 1.0f) + blin[0];
}

// ---------------------------------------------------------------------------
extern "C" void kernel_launch(void* const* d_in, const int* in_sizes, int n_in,
                              void* d_out, int out_size, void* d_ws,
                              size_t ws_size, hipStream_t stream) {
  const float* x    = (const float*)d_in[0];
  const float* W1   = (const float*)d_in[1];
  const float* b1   = (const float*)d_in[2];
  const float* W2   = (const float*)d_in[3];
  const float* b2   = (const float*)d_in[4];
  const float* W3   = (const float*)d_in[5];
  const float* b3   = (const float*)d_in[6];
  const float* Wlin = (const float*)d_in[7];
  const float* blin = (const float*)d_in[8];
  const int*   eidx = (const int*)d_in[9];
  const int*   bat  = (const int*)d_in[10];
  float* out = (float*)d_out;

  const int N = in_sizes[0] / 32;      // 100000
  const int E = in_sizes[9] / 2;       // 1600000
  const int G = out_size;              // 2048
  const int* row = eidx;
  const int* col = eidx + E;

  // workspace carve-out (all fp32)
  float* w       = (float*)d_ws;
  float* deg     = w;                 w += N;
  float* disqrt  = w;                 w += N;
  float* deginv  = w;                 w += N;
  float* nrm     = w;                 w += E;
  float* xw      = w;                 w += (size_t)N * HDIM;
  float* bufA    = w;                 w += (size_t)N * HDIM;
  float* bufB    = w;                 w += (size_t)N * HDIM;
  float* sums    = w;                 w += (size_t)G * HDIM;
  float* counts  = w;                 w += G;

  const int T = 256;
  const int gemm_blocks = (N + 63) / 64;
  const int edge_wblocks = (E + 7) / 8;

  // degrees & per-edge norm
  init_deg<<<(N + T - 1) / T, T, 0, stream>>>(deg, N);
  accum_deg<<<(E + T - 1) / T, T, 0, stream>>>(col, deg, E);
  finalize_deg<<<(N + T - 1) / T, T, 0, stream>>>(deg, disqrt, deginv, N);
  edge_norm<<<(E + T - 1) / T, T, 0, stream>>>(row, col, disqrt, nrm, E);

  // zero pooling accumulators (sums and counts are contiguous)
  zero_f32<<<(G * HDIM + G + T - 1) / T, T, 0, stream>>>(sums, G * HDIM + G);

  // ----- layer 1: h1 = relu(conv(x, W1, b1)) -----
  gcn_gemm_fused<32, false><<<gemm_blocks, T, 0, stream>>>(
      x, W1, b1, deginv, xw, bufA, N);
  edge_scatter<<<edge_wblocks, T, 0, stream>>>(xw, row, col, nrm, bufA, E);

  // ----- layer 2: h2 = relu(conv(h1, W2, b2)); relu(h1) fused in A-frag -----
  gcn_gemm_fused<128, true><<<gemm_blocks, T, 0, stream>>>(
      bufA, W2, b2, deginv, xw, bufB, N);
  edge_scatter<<<edge_wblocks, T, 0, stream>>>(xw, row, col, nrm, bufB, E);

  // ----- layer 3: h3 = conv(h2, W3, b3); relu(h2) fused in A-frag -----
  gcn_gemm_fused<128, true><<<gemm_blocks, T, 0, stream>>>(
      bufB, W3, b3, deginv, xw, bufA, N);
  edge_scatter<<<edge_wblocks, T, 0, stream>>>(xw, row, col, nrm, bufA, E);

  // ----- global mean pool + linear head -----
  pool_accum<<<(N + 7) / 8, T, 0, stream>>>(bufA, bat, sums, counts, N);
  final_pool<<<(G + 7) / 8, T, 0, stream>>>(sums, counts, Wlin, blin, out, G);
}